// LabelEncoder_27788438405708
// MI455X (gfx1250) — compile-verified
//
#include <hip/hip_runtime.h>

// Problem constants (from reference)
#define NNODES 10000
#define NEDGES 320000
#define IN_F   512
#define HID_F  256
#define OUT_F  128

// GEMM tiling
#define BK      64            // K-chunk staged in LDS per iteration
#define BN      128           // columns per block (8 waves * 16)
#define LDA     (BK + 4)      // A LDS stride: conflict-free (banks 4m+k+{0..3})
#define LDB     (BN + 16)     // B LDS stride: K+2 group lands +32 banks away

typedef __attribute__((ext_vector_type(2))) float v2f;
typedef __attribute__((ext_vector_type(8))) float v8f;

// ---------------------------------------------------------------------------
// fp32 atomic add -> native global_atomic_add_f32 on gfx1250
// ---------------------------------------------------------------------------
__device__ __forceinline__ void atomAddF(float* p, float v) {
  unsafeAtomicAdd(p, v);
}

// ---------------------------------------------------------------------------
// Degree kernels: deg[n] = 1 (self loop) + #edges with dst==n ; then rsqrt
// ---------------------------------------------------------------------------
__global__ void k_deg_init(float* deg) {
  int i = blockIdx.x * blockDim.x + threadIdx.x;
  if (i < NNODES) deg[i] = 1.0f;
}

__global__ void k_deg_count(const int* __restrict__ dst, float* deg) {
  int e = blockIdx.x * blockDim.x + threadIdx.x;
  if (e < NEDGES) atomAddF(&deg[dst[e]], 1.0f);
}

__global__ void k_deg_rsqrt(float* deg) {
  int i = blockIdx.x * blockDim.x + threadIdx.x;
  if (i < NNODES) deg[i] = rsqrtf(deg[i]);  // deg >= 1 always
}

// ---------------------------------------------------------------------------
// WMMA fp32 GEMM:  C[M,Nc] = A[M,K] * B[K,Nc]   (row-major, exact fp32)
//
// block = 256 threads (8 waves). Block tile = 16 rows x 128 cols.
// K is processed in BK=64 chunks: A chunk [16 x BK] and B chunk [BK x BN]
// are cooperatively staged into padded LDS (coalesced b128 global loads),
// then each wave runs 16 fully-unrolled V_WMMA_F32_16X16X4_F32 fed purely
// by immediate-offset ds reads. B is loaded once per *block*, not per wave.
//
// V_WMMA_F32_16X16X4_F32 VGPR layouts (ISA 7.12.2):
//   A 16x4 : lanes 0-15 -> (M=lane, K=k0..k0+1), lanes 16-31 -> K=k0+2..k0+3
//   B 4x16 : VGPR0: lanes0-15 (K=k0,N=lane), lanes16-31 (K=k0+2,N=lane-16)
//            VGPR1: K=k0+1 / K=k0+3
//   C/D    : VGPR j: lanes0-15 M=j, lanes16-31 M=j+8; N = lane&15
// ---------------------------------------------------------------------------
__global__ void k_gemm_wmma_f32(const float* __restrict__ A,
                                const float* __restrict__ B,
                                float* __restrict__ C,
                                int K, int Nc) {
  __shared__ float lA[16 * LDA];   // ~4.25 KB
  __shared__ float lB[BK * LDB];   // 36 KB   (WGP has 320 KB)

  const int tid  = threadIdx.x;
  const int lane = tid & 31;
  const int wave = tid >> 5;
  const int rowBase  = blockIdx.x * 16;
  const int colBlock = blockIdx.y * BN;

  const int m    = lane & 15;          // A row / C col-lane
  const int koff = (lane >> 4) << 1;   // 0 or 2
  const int nl   = (wave << 4) + (lane & 15);  // column within block tile

  v8f acc = {};

  for (int kb = 0; kb < K; kb += BK) {
    // --- stage A chunk [16 x BK]: 256 float4, one per thread ---
    {
      const int r  = tid >> 4;          // 16 float4 per row (BK/4)
      const int c4 = (tid & 15) << 2;
      float4 v = *reinterpret_cast<const float4*>(
          A + (size_t)(rowBase + r) * K + kb + c4);
      *reinterpret_cast<float4*>(lA + r * LDA + c4) = v;
    }
    // --- stage B chunk [BK x BN]: 2048 float4, 8 per thread ---
#pragma unroll
    for (int i = tid; i < (BK * BN) >> 2; i += 256) {
      const int r  = i >> 5;            // 32 float4 per row (BN/4)
      const int c4 = (i & 31) << 2;
      float4 v = *reinterpret_cast<const float4*>(
          B + (size_t)(kb + r) * Nc + colBlock + c4);
      *reinterpret_cast<float4*>(lB + r * LDB + c4) = v;
    }
    __syncthreads();

    // --- 16 WMMAs per chunk, all operands from LDS ---
#pragma unroll
    for (int k0 = 0; k0 < BK; k0 += 4) {
      v2f a, b;
      const float* ap = lA + m * LDA + (k0 + koff);   // ds_load_b64
      a.x = ap[0];
      a.y = ap[1];
      const float* bp = lB + (k0 + koff) * LDB + nl;  // 2x ds_load_b32
      b.x = bp[0];
      b.y = bp[LDB];
      acc = __builtin_amdgcn_wmma_f32_16x16x4_f32(
          /*neg_a=*/false, a, /*neg_b=*/false, b,
          /*c_mod=*/(short)0, acc, /*reuse_a=*/false, /*reuse_b=*/false);
    }
    __syncthreads();
  }

  // Store 16x16 tile per the C/D layout
  const int n = colBlock + nl;
#pragma unroll
  for (int j = 0; j < 8; ++j) {
    const int mm = j + ((lane >> 4) << 3);
    C[(size_t)(rowBase + mm) * Nc + n] = acc[j];
  }
}

// ---------------------------------------------------------------------------
// Self-loop init: out[n,f] = dinv[n]^2 * h[n,f]   (float4; F = 1<<fshift)
// ---------------------------------------------------------------------------
__global__ void k_self_init(const float* __restrict__ h,
                            const float* __restrict__ dinv,
                            float* __restrict__ out, int fshift) {
  int i = blockIdx.x * blockDim.x + threadIdx.x;  // float4 index
  int total = NNODES << (fshift - 2);
  if (i >= total) return;
  int node = i >> (fshift - 2);
  float s = dinv[node];
  s = s * s;
  float4 v = reinterpret_cast<const float4*>(h)[i];
  v.x *= s; v.y *= s; v.z *= s; v.w *= s;
  reinterpret_cast<float4*>(out)[i] = v;
}

// ---------------------------------------------------------------------------
// Edge scatter: out[dst] += dinv[src]*dinv[dst] * h[src]
// One wave per edge; float4 gathers; native f32 atomics into L2.
// ---------------------------------------------------------------------------
__global__ void k_edge_agg(const float* __restrict__ h,
                           const int* __restrict__ src,
                           const int* __restrict__ dst,
                           const float* __restrict__ dinv,
                           float* __restrict__ out, int F) {
  const int e = blockIdx.x * (blockDim.x >> 5) + (threadIdx.x >> 5);
  if (e >= NEDGES) return;
  const int lane = threadIdx.x & 31;
  const int s = src[e];
  const int d = dst[e];
  const float norm = dinv[s] * dinv[d];
  const float* __restrict__ hp = h + (size_t)s * F;
  float* op = out + (size_t)d * F;
  for (int f = lane << 2; f < F; f += 128) {
    float4 v = *reinterpret_cast<const float4*>(hp + f);
    atomAddF(op + f + 0, norm * v.x);
    atomAddF(op + f + 1, norm * v.y);
    atomAddF(op + f + 2, norm * v.z);
    atomAddF(op + f + 3, norm * v.w);
  }
}

// ---------------------------------------------------------------------------
// Bias (+ optional ReLU), in place. F is a power of two (256 / 128).
// ---------------------------------------------------------------------------
__global__ void k_bias_act(float* __restrict__ x, const float* __restrict__ b,
                           int fmask, int relu) {
  int i = blockIdx.x * blockDim.x + threadIdx.x;
  int total = NNODES * (fmask + 1);
  if (i >= total) return;
  float v = x[i] + b[i & fmask];
  if (relu) v = fmaxf(v, 0.0f);
  x[i] = v;
}

// ---------------------------------------------------------------------------
// Launcher
// ---------------------------------------------------------------------------
extern "C" void kernel_launch(void* const* d_in, const int* in_sizes, int n_in,
                              void* d_out, int out_size, void* d_ws, size_t ws_size,
                              hipStream_t stream) {
  const float* x   = (const float*)d_in[0];
  const int*   ei  = (const int*)d_in[1];   // [2, E]
  const float* W1  = (const float*)d_in[2];
  const float* b1  = (const float*)d_in[3];
  const float* W2  = (const float*)d_in[4];
  const float* b2  = (const float*)d_in[5];
  float* out = (float*)d_out;

  const int* src = ei;
  const int* dst = ei + NEDGES;

  char* ws = (char*)d_ws;
  float* deg = (float*)(ws);                                  // N floats (-> dinv)
  float* h1  = (float*)(ws + (64u << 10));                    // N*HID
  float* a1  = (float*)(ws + (64u << 10) + (size_t)NNODES * HID_F * 4);
  float* h2  = (float*)(ws + (64u << 10) + 2 * (size_t)NNODES * HID_F * 4);

  // 1) degrees -> dinv
  k_deg_init<<<(NNODES + 255) / 256, 256, 0, stream>>>(deg);
  k_deg_count<<<(NEDGES + 255) / 256, 256, 0, stream>>>(dst, deg);
  k_deg_rsqrt<<<(NNODES + 255) / 256, 256, 0, stream>>>(deg);

  // 2) layer 1: h1 = x @ W1   (M=10000=625*16, K=512, Nc=256)
  {
    dim3 grid(NNODES / 16, HID_F / BN);
    k_gemm_wmma_f32<<<grid, 256, 0, stream>>>(x, W1, h1, IN_F, HID_F);
  }
  // 3) propagate: a1 = Dinv (A+I) Dinv h1
  k_self_init<<<((NNODES * HID_F / 4) + 255) / 256, 256, 0, stream>>>(h1, deg, a1, 8);
  k_edge_agg<<<NEDGES / 8, 256, 0, stream>>>(h1, src, dst, deg, a1, HID_F);
  // 4) bias + ReLU
  k_bias_act<<<(NNODES * HID_F + 255) / 256, 256, 0, stream>>>(a1, b1, HID_F - 1, 1);

  // 5) layer 2: h2 = a1 @ W2  (K=256, Nc=128)
  {
    dim3 grid(NNODES / 16, OUT_F / BN);
    k_gemm_wmma_f32<<<grid, 256, 0, stream>>>(a1, W2, h2, HID_F, OUT_F);
  }
  // 6) propagate into d_out
  k_self_init<<<((NNODES * OUT_F / 4) + 255) / 256, 256, 0, stream>>>(h2, deg, out, 7);
  k_edge_agg<<<NEDGES / 8, 256, 0, stream>>>(h2, src, dst, deg, out, OUT_F);
  // 7) bias (no ReLU)
  k_bias_act<<<(NNODES * OUT_F + 255) / 256, 256, 0, stream>>>(out, b2, OUT_F - 1, 0);
}